// Net_21079699489113
// MI455X (gfx1250) — compile-verified
//
#include <hip/hip_runtime.h>
#include <math.h>

// ---------------- problem constants (match reference) ----------------
#define BG    16
#define NN    2048
#define EG    32768
#define FIN   32
#define HH    128
#define KK1   1639
#define KK2   1312
#define KK3   1050
#define MT1   128      // ceil(2048/16)
#define MT2   103      // ceil(1639/16) -> 1648 rows
#define MT3   82       // 1312/16

typedef __attribute__((ext_vector_type(2))) float v2f;
typedef __attribute__((ext_vector_type(8))) float v8f;

// ---------------------------------------------------------------------
// Fused GraphConv GEMM:  OUT = relu( X @ Wroot^T + AGG @ Wrel^T + bias )
// One wave32 computes a full 16(M) x 128(N) output strip:
//   - A-fragments (X row tile, AGG row tile) loaded ONCE per k-step
//   - reused across all 8 N-tiles -> 16 v_wmma_f32_16x16x4_f32 per k-step
// Fragment layouts per CDNA5 ISA 7.12.2:
//   A 16x4 (f32): lanes0-15 M=0..15 holding K={k0,k0+1}; lanes16-31 K={k0+2,k0+3}
//   B 4x16:       VGPR0 lanes0-15 K=0 / lanes16-31 K=2; VGPR1 K=1 / K=3
//   C/D 16x16:    VGPR r -> (M=r, N=lane) lo half ; (M=r+8, N=lane-16) hi half
// ---------------------------------------------------------------------
__global__ void gemm_conv_relu(const float* __restrict__ X,
                               const float* __restrict__ A,
                               const float* __restrict__ Wr,   // [128 x Fin]
                               const float* __restrict__ Wl,   // [128 x Fin]
                               const float* __restrict__ bias, // [128]
                               float* __restrict__ OUT,        // rows x 128
                               int Fin, int mtiles,
                               long xGS, long aGS, long oGS)
{
  int wave  = (int)((blockIdx.x * blockDim.x + threadIdx.x) >> 5);
  int lane  = threadIdx.x & 31;
  int total = BG * mtiles;
  if (wave >= total) return;                 // wave-uniform: EXEC stays all-1s

  int b  = wave / mtiles;
  int mt = wave % mtiles;

  const float* Xg = X + (size_t)b * xGS;
  const float* Ag = A + (size_t)b * aGS;

  int half = lane >> 4;        // 0 or 1
  int l15  = lane & 15;
  int row  = mt * 16 + l15;    // A-matrix M index (same in both half-waves)
  int koff = half * 2;         // K offset within the 4-wide step

  v8f c[8];
#pragma unroll
  for (int nt = 0; nt < 8; ++nt) {
    float bv = bias[nt * 16 + l15];
#pragma unroll
    for (int r = 0; r < 8; ++r) c[nt][r] = bv;
  }

  const float* xrow = Xg + (size_t)row * Fin;
  const float* arow = Ag + (size_t)row * Fin;

  for (int k0 = 0; k0 < Fin; k0 += 4) {
    v2f aX, aA;
    aX.x = xrow[k0 + koff];
    aX.y = xrow[k0 + koff + 1];
    aA.x = arow[k0 + koff];
    aA.y = arow[k0 + koff + 1];
#pragma unroll
    for (int nt = 0; nt < 8; ++nt) {
      int ncol = nt * 16 + l15;             // W row / output column
      v2f wr, wl;
      wr.x = Wr[(size_t)ncol * Fin + k0 + koff];
      wr.y = Wr[(size_t)ncol * Fin + k0 + koff + 1];
      wl.x = Wl[(size_t)ncol * Fin + k0 + koff];
      wl.y = Wl[(size_t)ncol * Fin + k0 + koff + 1];
      c[nt] = __builtin_amdgcn_wmma_f32_16x16x4_f32(false, aX, false, wr,
                                                    (short)0, c[nt], false, false);
      c[nt] = __builtin_amdgcn_wmma_f32_16x16x4_f32(false, aA, false, wl,
                                                    (short)0, c[nt], false, false);
    }
  }

  float* Og = OUT + (size_t)b * oGS;
#pragma unroll
  for (int nt = 0; nt < 8; ++nt) {
#pragma unroll
    for (int r = 0; r < 8; ++r) {
      float v = c[nt][r] > 0.f ? c[nt][r] : 0.f;   // fused ReLU
      int orow = mt * 16 + r + half * 8;
      Og[(size_t)orow * HH + nt * 16 + l15] = v;
    }
  }
}

// ------------------- scatter-add message passing ---------------------
// one block per (graph, edge); blockDim = F features
__global__ void scatter_add_k(const float* __restrict__ X, float* __restrict__ AGG,
                              const int* __restrict__ srcA, int sStride,
                              const int* __restrict__ dstA, int dStride,
                              const int* __restrict__ evA,
                              int F, long xGS, long aGS)
{
  int be = blockIdx.x;
  int b = be / EG, e = be % EG;
  if (evA && !evA[(size_t)b * EG + e]) return;
  int s = srcA[(size_t)b * sStride + e];
  int d = dstA[(size_t)b * dStride + e];
  int f = threadIdx.x;
  unsafeAtomicAdd(&AGG[(size_t)b * aGS + (size_t)d * F + f],
                  X[(size_t)b * xGS + (size_t)s * F + f]);
}

// ------------------------- tanh score ------------------------------
__global__ void score_k(const float* __restrict__ X, const float* __restrict__ p,
                        float* __restrict__ sc, int n, long GS)
{
  int b = blockIdx.y;
  int i = blockIdx.x * blockDim.x + threadIdx.x;
  if (i >= n) return;
  float ss = 0.f;
#pragma unroll 8
  for (int f = 0; f < HH; ++f) ss += p[f] * p[f];
  const float* xr = X + (size_t)b * GS + (size_t)i * HH;
  float d = 0.f;
#pragma unroll 8
  for (int f = 0; f < HH; ++f) d += xr[f] * p[f];
  sc[(size_t)b * NN + i] = tanhf(d / sqrtf(ss));
}

// ----------- per-graph bitonic top-k (descending) in LDS -------------
__global__ void topk_k(const float* __restrict__ sc, int n, int k,
                       int* __restrict__ perm, float* __restrict__ vals,
                       int* __restrict__ inv)
{
  __shared__ float sv[NN];
  __shared__ int   si[NN];
  int b = blockIdx.x, tid = threadIdx.x;
  for (int i = tid; i < NN; i += 256) {
    sv[i] = (i < n) ? sc[(size_t)b * NN + i] : -3.402823466e38f;
    si[i] = i;
  }
  __syncthreads();
  for (int kk = 2; kk <= NN; kk <<= 1) {
    for (int j = kk >> 1; j > 0; j >>= 1) {
      for (int i = tid; i < NN; i += 256) {
        int ixj = i ^ j;
        if (ixj > i) {
          bool up = ((i & kk) == 0);
          float vi = sv[i], vj = sv[ixj];
          bool sw = up ? (vi < vj) : (vi > vj);   // descending overall
          if (sw) {
            sv[i] = vj; sv[ixj] = vi;
            int t = si[i]; si[i] = si[ixj]; si[ixj] = t;
          }
        }
      }
      __syncthreads();
    }
  }
  for (int i = tid; i < NN; i += 256) inv[(size_t)b * NN + i] = -1;
  __syncthreads();
  for (int j = tid; j < k; j += 256) {
    perm[(size_t)b * NN + j] = si[j];
    vals[(size_t)b * NN + j] = sv[j];
    inv[(size_t)b * NN + si[j]] = j;
  }
}

// ------------------ gather x_new = x[perm] * vals --------------------
__global__ void gather_k(const float* __restrict__ Xin, const int* __restrict__ perm,
                         const float* __restrict__ vals, float* __restrict__ Xout,
                         int k, long GS)
{
  long idx = (long)blockIdx.x * blockDim.x + threadIdx.x;
  long total = (long)BG * k * HH;
  if (idx >= total) return;
  int b = (int)(idx / ((long)k * HH));
  int r = (int)(idx % ((long)k * HH));
  int j = r / HH, f = r % HH;
  int pj = perm[(size_t)b * NN + j];
  Xout[(size_t)b * GS + (size_t)j * HH + f] =
      Xin[(size_t)b * GS + (size_t)pj * HH + f] * vals[(size_t)b * NN + j];
}

// --------------------- edge remap after pooling ----------------------
__global__ void remap_k(const int* __restrict__ srcIn, int sStride,
                        const int* __restrict__ dstIn, int dStride,
                        const int* __restrict__ evIn, const int* __restrict__ inv,
                        int* __restrict__ srcOut, int* __restrict__ dstOut,
                        int* __restrict__ evOut)
{
  long idx = (long)blockIdx.x * blockDim.x + threadIdx.x;
  if (idx >= (long)BG * EG) return;
  int b = (int)(idx / EG), e = (int)(idx % EG);
  int s = srcIn[(size_t)b * sStride + e];
  int d = dstIn[(size_t)b * dStride + e];
  int v = evIn ? evIn[(size_t)b * EG + e] : 1;
  int ns = inv[(size_t)b * NN + s];
  int nd = inv[(size_t)b * NN + d];
  int ok = (v && ns >= 0 && nd >= 0) ? 1 : 0;
  srcOut[(size_t)b * EG + e] = ok ? ns : 0;
  dstOut[(size_t)b * EG + e] = ok ? nd : 0;
  evOut[(size_t)b * EG + e]  = ok;
}

// -------------------- readout: [max ; mean] accumulate ---------------
__global__ void readout_k(const float* __restrict__ X, float* __restrict__ racc,
                          int k, long GS)
{
  int b = blockIdx.x, f = threadIdx.x;
  float m = -3.402823466e38f, s = 0.f;
  for (int j = 0; j < k; ++j) {
    float v = X[(size_t)b * GS + (size_t)j * HH + f];
    m = v > m ? v : m;
    s += v;
  }
  racc[b * 256 + f]       += m;
  racc[b * 256 + 128 + f] += s / (float)k;
}

// --------------------------- MLP head --------------------------------
__global__ void mlp_k(const float* __restrict__ racc,
                      const float* __restrict__ W1, const float* __restrict__ b1,
                      const float* __restrict__ W2, const float* __restrict__ b2,
                      const float* __restrict__ W3, const float* __restrict__ b3,
                      float* __restrict__ out)
{
  __shared__ float g[256], h1[128], h2[64];
  int tid = threadIdx.x;
  for (int b = 0; b < BG; ++b) {
    g[tid] = racc[b * 256 + tid];
    __syncthreads();
    if (tid < 128) {
      float a = b1[tid];
      for (int i = 0; i < 256; ++i) a += W1[tid * 256 + i] * g[i];
      h1[tid] = a > 0.f ? a : 0.f;
    }
    __syncthreads();
    if (tid < 64) {
      float a = b2[tid];
      for (int i = 0; i < 128; ++i) a += W2[tid * 128 + i] * h1[i];
      h2[tid] = a > 0.f ? a : 0.f;
    }
    __syncthreads();
    if (tid == 0) {
      float a = b3[0];
      for (int i = 0; i < 64; ++i) a += W3[i] * h2[i];
      out[b] = 1.f / (1.f + expf(-a));
    }
    __syncthreads();
  }
}

// =====================================================================
extern "C" void kernel_launch(void* const* d_in, const int* in_sizes, int n_in,
                              void* d_out, int out_size, void* d_ws, size_t ws_size,
                              hipStream_t stream)
{
  const float* x      = (const float*)d_in[0];
  const int*   ei     = (const int*)d_in[1];
  const float* Wroot1 = (const float*)d_in[2];
  const float* Wrel1  = (const float*)d_in[3];
  const float* bb1    = (const float*)d_in[4];
  const float* p1     = (const float*)d_in[5];
  const float* Wroot2 = (const float*)d_in[6];
  const float* Wrel2  = (const float*)d_in[7];
  const float* bb2    = (const float*)d_in[8];
  const float* p2     = (const float*)d_in[9];
  const float* Wroot3 = (const float*)d_in[10];
  const float* Wrel3  = (const float*)d_in[11];
  const float* bb3    = (const float*)d_in[12];
  const float* p3     = (const float*)d_in[13];
  const float* lin1W  = (const float*)d_in[14];
  const float* lin1b  = (const float*)d_in[15];
  const float* lin2W  = (const float*)d_in[16];
  const float* lin2b  = (const float*)d_in[17];
  const float* lin3W  = (const float*)d_in[18];
  const float* lin3b  = (const float*)d_in[19];
  float* out = (float*)d_out;

  // ---- workspace layout (floats) ----
  const size_t SB = (size_t)BG * NN * HH;          // 4 Mi elems = 16 MiB
  float* hbuf  = (float*)d_ws;
  float* xpool = hbuf + SB;
  float* agg   = xpool + SB;
  int*   esrc  = (int*)(agg + SB);
  int*   edst  = esrc + (size_t)BG * EG;
  int*   eev   = edst + (size_t)BG * EG;
  float* scores = (float*)(eev + (size_t)BG * EG);
  float* vals   = scores + (size_t)BG * NN;
  int*   perm   = (int*)(vals + (size_t)BG * NN);
  int*   inv    = perm + (size_t)BG * NN;
  float* racc   = (float*)(inv + (size_t)BG * NN);

  const long GS = (long)NN * HH;                   // 2048*128 per-graph stride

  hipMemsetAsync(racc, 0, BG * 256 * sizeof(float), stream);

  // =================== Layer 1 (Fin=32, n=2048) =====================
  hipMemsetAsync(agg, 0, SB * sizeof(float), stream);
  scatter_add_k<<<BG * EG, FIN, 0, stream>>>(x, agg, ei, 2 * EG, ei + EG, 2 * EG,
                                             nullptr, FIN, (long)NN * FIN,
                                             (long)NN * FIN);
  {
    int waves = BG * MT1;
    gemm_conv_relu<<<(waves + 3) / 4, 128, 0, stream>>>(
        x, agg, Wroot1, Wrel1, bb1, hbuf, FIN, MT1,
        (long)NN * FIN, (long)NN * FIN, GS);
  }
  score_k<<<dim3((NN + 255) / 256, BG), 256, 0, stream>>>(hbuf, p1, scores, NN, GS);
  topk_k<<<BG, 256, 0, stream>>>(scores, NN, KK1, perm, vals, inv);
  gather_k<<<(int)(((long)BG * KK1 * HH + 255) / 256), 256, 0, stream>>>(
      hbuf, perm, vals, xpool, KK1, GS);
  readout_k<<<BG, HH, 0, stream>>>(xpool, racc, KK1, GS);
  remap_k<<<(BG * EG + 255) / 256, 256, 0, stream>>>(ei, 2 * EG, ei + EG, 2 * EG,
                                                     nullptr, inv, esrc, edst, eev);

  // =================== Layer 2 (Fin=128, n=K1) ======================
  hipMemsetAsync(agg, 0, SB * sizeof(float), stream);
  scatter_add_k<<<BG * EG, HH, 0, stream>>>(xpool, agg, esrc, EG, edst, EG, eev,
                                            HH, GS, GS);
  {
    int waves = BG * MT2;
    gemm_conv_relu<<<(waves + 3) / 4, 128, 0, stream>>>(
        xpool, agg, Wroot2, Wrel2, bb2, hbuf, HH, MT2, GS, GS, GS);
  }
  score_k<<<dim3((KK1 + 255) / 256, BG), 256, 0, stream>>>(hbuf, p2, scores, KK1, GS);
  topk_k<<<BG, 256, 0, stream>>>(scores, KK1, KK2, perm, vals, inv);
  gather_k<<<(int)(((long)BG * KK2 * HH + 255) / 256), 256, 0, stream>>>(
      hbuf, perm, vals, xpool, KK2, GS);
  readout_k<<<BG, HH, 0, stream>>>(xpool, racc, KK2, GS);
  remap_k<<<(BG * EG + 255) / 256, 256, 0, stream>>>(esrc, EG, edst, EG, eev, inv,
                                                     esrc, edst, eev);

  // =================== Layer 3 (Fin=128, n=K2) ======================
  hipMemsetAsync(agg, 0, SB * sizeof(float), stream);
  scatter_add_k<<<BG * EG, HH, 0, stream>>>(xpool, agg, esrc, EG, edst, EG, eev,
                                            HH, GS, GS);
  {
    int waves = BG * MT3;
    gemm_conv_relu<<<(waves + 3) / 4, 128, 0, stream>>>(
        xpool, agg, Wroot3, Wrel3, bb3, hbuf, HH, MT3, GS, GS, GS);
  }
  score_k<<<dim3((KK2 + 255) / 256, BG), 256, 0, stream>>>(hbuf, p3, scores, KK2, GS);
  topk_k<<<BG, 256, 0, stream>>>(scores, KK2, KK3, perm, vals, inv);
  gather_k<<<(int)(((long)BG * KK3 * HH + 255) / 256), 256, 0, stream>>>(
      hbuf, perm, vals, xpool, KK3, GS);
  readout_k<<<BG, HH, 0, stream>>>(xpool, racc, KK3, GS);

  // ========================== MLP head ==============================
  mlp_k<<<1, 256, 0, stream>>>(racc, lin1W, lin1b, lin2W, lin2b, lin3W, lin3b, out);
}